// AE_gnnrnn_40364102648138
// MI455X (gfx1250) — compile-verified
//
#include <hip/hip_runtime.h>

typedef __attribute__((ext_vector_type(2))) float v2f;
typedef __attribute__((ext_vector_type(8))) float v8f;

#define DEVFN __device__ __forceinline__

constexpr int NN  = 128;   // nodes
constexpr int LL  = 128;   // seq len
constexpr int VV  = 8000;  // vocab
constexpr int IND = 64;    // embedding dim
constexpr int HH  = 13;    // LSTM hidden
constexpr int GG  = 52;    // 4*H gates
constexpr int H2  = 26;    // 2*H

DEVFN float sigf(float x) { return 1.0f / (1.0f + __expf(-x)); }

// ---------------------------------------------------------------- utilities
__global__ void k_fill0(float* p, int n) {
    int i = blockIdx.x * blockDim.x + threadIdx.x;
    if (i < n) p[i] = 0.0f;
}

// generic small dense: out[m][o] = act( sum_k in[m,k]*W(k,o) + b[o] )
// transW=1 -> W is [Nc,K] row-major used transposed (x @ W.T), else W is [K,Nc]
__global__ void k_dense(const float* __restrict__ in, const float* __restrict__ W,
                        const float* __restrict__ b, float* __restrict__ out,
                        int M, int K, int Nc, int transW, int leaky) {
    int idx = blockIdx.x * blockDim.x + threadIdx.x;
    if (idx >= M * Nc) return;
    int m = idx / Nc, o = idx - m * Nc;
    float s = b ? b[o] : 0.0f;
    for (int k = 0; k < K; ++k)
        s += in[m * K + k] * (transW ? W[o * K + k] : W[k * Nc + o]);
    if (leaky) s = (s > 0.0f) ? s : 0.01f * s;
    out[idx] = s;
}

// ---------------------------------------------------------------- encoder biLSTM
// one block (64 threads) per (node, direction); serial over L steps.
__global__ __launch_bounds__(64) void k_lstm_scan(
    const int* __restrict__ tok, const float* __restrict__ emb,
    const float* __restrict__ Wih_f, const float* __restrict__ Whh_f, const float* __restrict__ b_f,
    const float* __restrict__ Wih_b, const float* __restrict__ Whh_b, const float* __restrict__ b_b,
    float* __restrict__ Hcat, float* __restrict__ Ccat) {
    int n = blockIdx.x >> 1;
    int dir = blockIdx.x & 1;
    const float* Wih = dir ? Wih_b : Wih_f;   // [52,64]
    const float* Whh = dir ? Whh_b : Whh_f;   // [52,13]
    const float* bb  = dir ? b_b  : b_f;      // [52]

    __shared__ float sWihT[IND * GG];         // transposed: [d][g] to avoid bank conflicts
    __shared__ float x[IND];
    __shared__ float h[16], c[16];
    __shared__ float gate[GG];

    int t0 = threadIdx.x;
    for (int idx = t0; idx < GG * IND; idx += 64) {
        int g = idx / IND, d = idx - g * IND;
        sWihT[d * GG + g] = Wih[idx];
    }
    if (t0 < 16) { h[t0] = 0.0f; c[t0] = 0.0f; }

    float whh[HH];
    float bg = 0.0f;
    if (t0 < GG) {
#pragma unroll
        for (int j = 0; j < HH; ++j) whh[j] = Whh[t0 * HH + j];
        bg = bb[t0];
    }
    __syncthreads();

    for (int t = 0; t < LL; ++t) {
        int l = dir ? (LL - 1 - t) : t;
        int token = tok[n * LL + l];
        x[t0] = emb[token * IND + t0];        // 64 threads load the 64-wide embedding row
        __syncthreads();
        if (t0 < GG) {
            float g = bg;
#pragma unroll 8
            for (int d = 0; d < IND; ++d) g += sWihT[d * GG + t0] * x[d];
#pragma unroll
            for (int j = 0; j < HH; ++j) g += whh[j] * h[j];
            gate[t0] = g;
        }
        __syncthreads();
        if (t0 < HH) {
            float ii = sigf(gate[t0]);
            float ff = sigf(gate[HH + t0]);
            float gg = tanhf(gate[2 * HH + t0]);
            float oo = sigf(gate[3 * HH + t0]);
            float cn = ff * c[t0] + ii * gg;
            c[t0] = cn;
            h[t0] = oo * tanhf(cn);
        }
        __syncthreads();
    }
    if (t0 < HH) {
        Hcat[n * H2 + dir * HH + t0] = h[t0];
        Ccat[n * H2 + dir * HH + t0] = c[t0];
    }
}

// ---------------------------------------------------------------- GCN pieces
__global__ void k_deg_init(float* deg) {
    int i = blockIdx.x * blockDim.x + threadIdx.x;
    if (i < NN) deg[i] = 1.0f;                // self-loop
}
__global__ void k_deg_accum(const int* __restrict__ edge, float* deg, int E) {
    int e = blockIdx.x * blockDim.x + threadIdx.x;
    if (e < E) atomicAdd(&deg[edge[E + e]], 1.0f);   // dst row
}
__global__ void k_deg_finish(float* deg) {
    int i = blockIdx.x * blockDim.x + threadIdx.x;
    if (i < NN) deg[i] = rsqrtf(fmaxf(deg[i], 1.0f));
}
// msg scatter: out[dst] += dinv[src]*dinv[dst]*y[src]; edges [0,E) real, [E,E+N) self loops
__global__ void k_gcn_scatter(const float* __restrict__ y, const int* __restrict__ edge,
                              const float* __restrict__ dinv, float* __restrict__ out,
                              int E, int F) {
    int idx = blockIdx.x * blockDim.x + threadIdx.x;
    int tot = (E + NN) * F;
    if (idx >= tot) return;
    int e = idx / F, f = idx - e * F;
    int s, d;
    if (e < E) { s = edge[e]; d = edge[E + e]; } else { s = d = e - E; }
    atomicAdd(&out[d * F + f], dinv[s] * dinv[d] * y[s * F + f]);
}
__global__ void k_bias_leaky(float* __restrict__ x, const float* __restrict__ b, int M, int F) {
    int idx = blockIdx.x * blockDim.x + threadIdx.x;
    if (idx >= M * F) return;
    float v = x[idx] + b[idx % F];
    x[idx] = (v > 0.0f) ? v : 0.01f * v;
}

// ---------------------------------------------------------------- decoder cell
// one thread per (n,l,j), j<13; handles both directions.
__global__ void k_decoder(const int* __restrict__ tok,
                          const float* __restrict__ sh, const float* __restrict__ sc,
                          const float* __restrict__ Wih_f, const float* __restrict__ Whh_f, const float* __restrict__ b_f,
                          const float* __restrict__ Wih_b, const float* __restrict__ Whh_b, const float* __restrict__ b_b,
                          float* __restrict__ dec_out) {
    int idx = blockIdx.x * blockDim.x + threadIdx.x;
    if (idx >= NN * LL * HH) return;
    int j = idx % HH;
    int nl = idx / HH;
    int l = nl % LL, n = nl / LL;
    float prev = (l == 0) ? -1.0f : (float)tok[n * LL + l - 1];
    for (int dir = 0; dir < 2; ++dir) {
        const float* Wih = dir ? Wih_b : Wih_f;  // [52,1]
        const float* Whh = dir ? Whh_b : Whh_f;  // [52,13]
        const float* bb  = dir ? b_b  : b_f;
        float g[4];
#pragma unroll
        for (int q = 0; q < 4; ++q) {
            int gi = q * HH + j;
            float s = prev * Wih[gi] + bb[gi];
            if (l == 0) {
#pragma unroll
                for (int k = 0; k < HH; ++k) s += Whh[gi * HH + k] * sh[n * H2 + dir * HH + k];
            }
            g[q] = s;
        }
        float c0 = (l == 0) ? sc[n * H2 + dir * HH + j] : 0.0f;
        float cc = sigf(g[1]) * c0 + sigf(g[0]) * tanhf(g[2]);
        float hh = sigf(g[3]) * tanhf(cc);
        dec_out[(n * LL + l) * H2 + dir * HH + j] = hh;
    }
}

// ---------------------------------------------------------------- output GEMM (WMMA f32)
// out[m, v] = sum_k dec[m,k]*Wout[v,k] + bout[v]; M=16384, K=26 (pad 28), V=8000
// 4 waves/block; each wave owns one 16x16 tile: 7x V_WMMA_F32_16X16X4_F32.
// Tail chunk (K=24..27) handled branch-free: only lower-half lanes (khalf==0,
// i.e. K=24,25) carry data; upper-half lanes (K=26,27) are zeroed by a mask.
__global__ __launch_bounds__(128) void k_out_gemm(const float* __restrict__ dec,
                                                  const float* __restrict__ Wout,
                                                  const float* __restrict__ bout,
                                                  float* __restrict__ out) {
    const int wave = threadIdx.x >> 5;
    const int lane = threadIdx.x & 31;
    const int row0 = blockIdx.x * 16;                 // M tile (1024 tiles)
    const int ntile = blockIdx.y * 4 + wave;          // 500 N tiles total
    const int col0 = ntile * 16;

    const int m16 = lane & 15;                        // M (for A) / N (for B,C)
    const int khalf = (lane >> 4) * 2;                // K sub-offset for A and B lanes

    const float* arow = dec  + (size_t)(row0 + m16) * H2;
    const float* brow = Wout + (size_t)(col0 + m16) * H2;

    v8f acc = {};
#pragma unroll
    for (int kc = 0; kc < 6; ++kc) {                  // K = 0..23, no predication
        int kb = kc * 4;
        v2f a, b;
#pragma unroll
        for (int v = 0; v < 2; ++v) {
            int k = kb + khalf + v;
            a[v] = arow[k];
            b[v] = brow[k];
        }
        acc = __builtin_amdgcn_wmma_f32_16x16x4_f32(false, a, false, b,
                                                    (short)0, acc, false, false);
    }
    {                                                 // K = 24..27 tail, branch-free
        float msk = (khalf == 0) ? 1.0f : 0.0f;
        v2f a, b;
#pragma unroll
        for (int v = 0; v < 2; ++v) {                 // addresses 24,25 always in-bounds
            a[v] = msk * arow[24 + v];
            b[v] = msk * brow[24 + v];
        }
        acc = __builtin_amdgcn_wmma_f32_16x16x4_f32(false, a, false, b,
                                                    (short)0, acc, false, false);
    }
    float bias = bout[col0 + m16];
#pragma unroll
    for (int r = 0; r < 8; ++r) {
        int m = row0 + r + (lane >> 4) * 8;
        out[(size_t)m * VV + col0 + m16] = acc[r] + bias;
    }
}

// ---------------------------------------------------------------- launch
extern "C" void kernel_launch(void* const* d_in, const int* in_sizes, int n_in,
                              void* d_out, int out_size, void* d_ws, size_t ws_size,
                              hipStream_t stream) {
    const int*   tok   = (const int*)d_in[0];
    const int*   edge  = (const int*)d_in[1];
    const float* emb   = (const float*)d_in[2];
    const float* Wih_f = (const float*)d_in[3];
    const float* Whh_f = (const float*)d_in[4];
    const float* b_f   = (const float*)d_in[5];
    const float* Wih_b = (const float*)d_in[6];
    const float* Whh_b = (const float*)d_in[7];
    const float* b_b   = (const float*)d_in[8];
    const float* Wp1   = (const float*)d_in[9];
    const float* bp1   = (const float*)d_in[10];
    const float* Wp2   = (const float*)d_in[11];
    const float* bp2   = (const float*)d_in[12];
    const float* gW1[2] = {(const float*)d_in[13], (const float*)d_in[19]};
    const float* gb1[2] = {(const float*)d_in[14], (const float*)d_in[20]};
    const float* gW2[2] = {(const float*)d_in[15], (const float*)d_in[21]};
    const float* gb2[2] = {(const float*)d_in[16], (const float*)d_in[22]};
    const float* gWf[2] = {(const float*)d_in[17], (const float*)d_in[23]};
    const float* gbf[2] = {(const float*)d_in[18], (const float*)d_in[24]};
    const float* Wih_fd = (const float*)d_in[25];
    const float* Whh_fd = (const float*)d_in[26];
    const float* b_fd   = (const float*)d_in[27];
    const float* Wih_bd = (const float*)d_in[28];
    const float* Whh_bd = (const float*)d_in[29];
    const float* b_bd   = (const float*)d_in[30];
    const float* Wout   = (const float*)d_in[31];
    const float* bout   = (const float*)d_in[32];
    float* out = (float*)d_out;

    const int E = in_sizes[1] / 2;  // 2048

    // workspace carve-up (floats)
    float* ws = (float*)d_ws;
    float* Hcat   = ws;                 // 128*26
    float* Ccat   = Hcat + NN * H2;
    float* st_h   = Ccat + NN * H2;     // state_h
    float* st_c   = st_h + NN * H2;
    float* dinv   = st_c + NN * H2;     // 128
    float* y1     = dinv + NN;          // 128*16
    float* a1     = y1 + NN * 16;       // 128*16
    float* y2     = a1 + NN * 16;       // 128*32
    float* a2     = y2 + NN * 32;       // 128*32
    float* s_out[2];
    s_out[0]      = a2 + NN * 32;       // sh 128*26
    s_out[1]      = s_out[0] + NN * H2; // sc 128*26
    float* decb   = s_out[1] + NN * H2; // dec_out 128*128*26

    // 1) encoder biLSTM
    k_lstm_scan<<<NN * 2, 64, 0, stream>>>(tok, emb, Wih_f, Whh_f, b_f,
                                           Wih_b, Whh_b, b_b, Hcat, Ccat);
    // 2) state projections (x @ W.T + b)
    k_dense<<<(NN * H2 + 255) / 256, 256, 0, stream>>>(Hcat, Wp1, bp1, st_h, NN, H2, H2, 1, 0);
    k_dense<<<(NN * H2 + 255) / 256, 256, 0, stream>>>(Ccat, Wp2, bp2, st_c, NN, H2, H2, 1, 0);
    // 3) symmetric-norm degrees
    k_deg_init<<<1, NN, 0, stream>>>(dinv);
    k_deg_accum<<<(E + 255) / 256, 256, 0, stream>>>(edge, dinv, E);
    k_deg_finish<<<1, NN, 0, stream>>>(dinv);
    // 4) two GCN stacks (gh on state_h, gc on state_c)
    const float* xin[2] = {st_h, st_c};
    for (int p = 0; p < 2; ++p) {
        k_dense<<<(NN * 16 + 255) / 256, 256, 0, stream>>>(xin[p], gW1[p], nullptr, y1, NN, H2, 16, 0, 0);
        k_fill0<<<(NN * 16 + 255) / 256, 256, 0, stream>>>(a1, NN * 16);
        k_gcn_scatter<<<((E + NN) * 16 + 255) / 256, 256, 0, stream>>>(y1, edge, dinv, a1, E, 16);
        k_bias_leaky<<<(NN * 16 + 255) / 256, 256, 0, stream>>>(a1, gb1[p], NN, 16);

        k_dense<<<(NN * 32 + 255) / 256, 256, 0, stream>>>(a1, gW2[p], nullptr, y2, NN, 16, 32, 0, 0);
        k_fill0<<<(NN * 32 + 255) / 256, 256, 0, stream>>>(a2, NN * 32);
        k_gcn_scatter<<<((E + NN) * 32 + 255) / 256, 256, 0, stream>>>(y2, edge, dinv, a2, E, 32);
        k_bias_leaky<<<(NN * 32 + 255) / 256, 256, 0, stream>>>(a2, gb2[p], NN, 32);

        k_dense<<<(NN * H2 + 255) / 256, 256, 0, stream>>>(a2, gWf[p], gbf[p], s_out[p], NN, 32, H2, 0, 0);
    }
    // 5) decoder single-step biLSTM cells
    k_decoder<<<(NN * LL * HH + 255) / 256, 256, 0, stream>>>(tok, s_out[0], s_out[1],
                                                              Wih_fd, Whh_fd, b_fd,
                                                              Wih_bd, Whh_bd, b_bd, decb);
    // 6) big output GEMM with f32 WMMA: [16384,26] x [26,8000]
    dim3 grid((NN * LL) / 16, (VV / 16) / 4);   // (1024, 125), 4 waves/block
    k_out_gemm<<<grid, 128, 0, stream>>>(decb, Wout, bout, out);
}